// QNN_31155692765447
// MI455X (gfx1250) — compile-verified
//
#include <hip/hip_runtime.h>
#include <hip/hip_bf16.h>

typedef float v2f __attribute__((ext_vector_type(2)));
typedef float v4f __attribute__((ext_vector_type(4)));
typedef float v8f __attribute__((ext_vector_type(8)));

// ---------------------------------------------------------------------------
// Stage 1: build per-parameter-row 16x16 complex unitaries and store them as
// WMMA A-fragments (U in the 16x16x4-f32 A lane layout:
//   lanes 0-15: (M=lane, K=v), lanes 16-31: (M=lane-16, K=v+2)).
// One block (one wave32) per parameter row p; U kept in LDS.
// ---------------------------------------------------------------------------

__device__ const int QNN_CRX_C[12] = {0,0,0,1,1,1,2,2,2,3,3,3};
__device__ const int QNN_CRX_T[12] = {1,2,3,0,2,3,0,1,3,0,1,2};

// U <- G * U for a 1-qubit gate [[a,b],[c,d]] (complex) on wire w.
__device__ __forceinline__ void qnn_apply1q(float (*Ur)[16], float (*Ui)[16], int w,
    float arr, float ari, float brr, float bri,
    float crr, float cri, float drr, float dri, int t)
{
  const int m = 8 >> w;                 // wire w -> bit (3-w)
  for (int task = t; task < 128; task += 32) {
    const int j  = task & 15;           // column
    const int pi = task >> 4;           // row-pair index
    const int r0 = ((pi & ~(m - 1)) << 1) | (pi & (m - 1));
    const int r1 = r0 | m;
    const float x0r = Ur[r0][j], x0i = Ui[r0][j];
    const float x1r = Ur[r1][j], x1i = Ui[r1][j];
    Ur[r0][j] = arr*x0r - ari*x0i + brr*x1r - bri*x1i;
    Ui[r0][j] = arr*x0i + ari*x0r + brr*x1i + bri*x1r;
    Ur[r1][j] = crr*x0r - cri*x0i + drr*x1r - dri*x1i;
    Ui[r1][j] = crr*x0i + cri*x0r + drr*x1i + dri*x1r;
  }
  __syncthreads();
}

// U <- G * U for a 2-qubit gate (4x4 complex, row index = b_w1*2 + b_w2).
__device__ __forceinline__ void qnn_apply2q(float (*Ur)[16], float (*Ui)[16],
    int w1, int w2, const float (&Mr)[4][4], const float (&Mi)[4][4], int t)
{
  const int m1 = 8 >> w1, m2 = 8 >> w2;
  const int mlo = (m1 < m2) ? m1 : m2;
  const int mhi = m1 ^ m2 ^ mlo;
  for (int task = t; task < 64; task += 32) {
    const int j = task & 15;
    const int q = task >> 4;            // 2 free bits
    const int t1 = ((q  & ~(mlo - 1)) << 1) | (q  & (mlo - 1));
    const int r  = ((t1 & ~(mhi - 1)) << 1) | (t1 & (mhi - 1));
    const int rows[4] = { r, r | m2, r | m1, r | m1 | m2 };
    float xr[4], xi[4];
    #pragma unroll
    for (int c = 0; c < 4; ++c) { xr[c] = Ur[rows[c]][j]; xi[c] = Ui[rows[c]][j]; }
    #pragma unroll
    for (int o = 0; o < 4; ++o) {
      float sr = 0.f, si = 0.f;
      #pragma unroll
      for (int c = 0; c < 4; ++c) {
        sr += Mr[o][c]*xr[c] - Mi[o][c]*xi[c];
        si += Mr[o][c]*xi[c] + Mi[o][c]*xr[c];
      }
      Ur[rows[o]][j] = sr; Ui[rows[o]][j] = si;
    }
  }
  __syncthreads();
}

__global__ __launch_bounds__(32) void qnn_build_kernel(
    const float* __restrict__ p_rotation,   // [16,12]
    const float* __restrict__ p_ising,      // [16,4]
    const float* __restrict__ p_entangle,   // [16,12]
    float* __restrict__ bfrag)              // [16][32 lanes][16 floats]
{
  __shared__ float Ur[16][16];
  __shared__ float Ui[16][16];
  const int p = blockIdx.x;
  const int t = threadIdx.x;

  for (int e = t; e < 256; e += 32) {
    const int r = e >> 4, c = e & 15;
    Ur[r][c] = (r == c) ? 1.f : 0.f;
    Ui[r][c] = 0.f;
  }
  __syncthreads();

  float cth, sth;
  // layer 1: RY(p_rotation[p, i]) on wire i
  for (int i = 0; i < 4; ++i) {
    __sincosf(p_rotation[p*12 + i] * 0.5f, &sth, &cth);
    qnn_apply1q(Ur, Ui, i, cth,0.f, -sth,0.f, sth,0.f, cth,0.f, t);
  }
  // IsingXX on (0,1) and (2,3): c*I4 - i*s*antidiag
  {
    const int w1[2] = {0, 2}, w2[2] = {1, 3};
    for (int idx = 0; idx < 2; ++idx) {
      __sincosf(p_ising[p*4 + idx] * 0.5f, &sth, &cth);
      const float Mr[4][4] = {{cth,0,0,0},{0,cth,0,0},{0,0,cth,0},{0,0,0,cth}};
      const float Mi[4][4] = {{0,0,0,-sth},{0,0,-sth,0},{0,-sth,0,0},{-sth,0,0,0}};
      qnn_apply2q(Ur, Ui, w1[idx], w2[idx], Mr, Mi, t);
    }
  }
  // layer 2: RX(p_rotation[p, 4+i]) on wire i
  for (int i = 0; i < 4; ++i) {
    __sincosf(p_rotation[p*12 + 4 + i] * 0.5f, &sth, &cth);
    qnn_apply1q(Ur, Ui, i, cth,0.f, 0.f,-sth, 0.f,-sth, cth,0.f, t);
  }
  // IsingXX on (1,2) and (3,0)
  {
    const int w1[2] = {1, 3}, w2[2] = {2, 0};
    for (int idx = 0; idx < 2; ++idx) {
      __sincosf(p_ising[p*4 + 2 + idx] * 0.5f, &sth, &cth);
      const float Mr[4][4] = {{cth,0,0,0},{0,cth,0,0},{0,0,cth,0},{0,0,0,cth}};
      const float Mi[4][4] = {{0,0,0,-sth},{0,0,-sth,0},{0,-sth,0,0},{-sth,0,0,0}};
      qnn_apply2q(Ur, Ui, w1[idx], w2[idx], Mr, Mi, t);
    }
  }
  // layer 3: RY(p_rotation[p, 8+i]) on wire i
  for (int i = 0; i < 4; ++i) {
    __sincosf(p_rotation[p*12 + 8 + i] * 0.5f, &sth, &cth);
    qnn_apply1q(Ur, Ui, i, cth,0.f, -sth,0.f, sth,0.f, cth,0.f, t);
  }
  // CRX entangling layer: diag(1,1, RX)
  for (int idx = 0; idx < 12; ++idx) {
    __sincosf(p_entangle[p*12 + idx] * 0.5f, &sth, &cth);
    const float Mr[4][4] = {{1,0,0,0},{0,1,0,0},{0,0,cth,0},{0,0,0,cth}};
    const float Mi[4][4] = {{0,0,0,0},{0,0,0,0},{0,0,0,-sth},{0,0,-sth,0}};
    qnn_apply2q(Ur, Ui, QNN_CRX_C[idx], QNN_CRX_T[idx], Mr, Mi, t);
  }

  // Emit U as A-fragments. Per lane: 16 floats, per kc the b128-friendly
  // order { Ur(K=v0), Ur(K=v1), Ui(K=v0), Ui(K=v1) } so stage 2 gets both
  // v2f operands contiguously from one 128-bit load (no repacking movs).
  const int n  = t & 15;
  const int hi = t >> 4;
  #pragma unroll
  for (int kc = 0; kc < 4; ++kc) {
    const int k0 = 4*kc + 2*hi;
    bfrag[p*512 + t*16 + kc*4 + 0] = Ur[n][k0 + 0];
    bfrag[p*512 + t*16 + kc*4 + 1] = Ur[n][k0 + 1];
    bfrag[p*512 + t*16 + kc*4 + 2] = Ui[n][k0 + 0];
    bfrag[p*512 + t*16 + kc*4 + 3] = Ui[n][k0 + 1];
  }
}

// ---------------------------------------------------------------------------
// Stage 2: per 16-sample tile, complex GEMM  phi^T = U * psi^T  via
// v_wmma_f32_16x16x4_f32 with A = U fragments, B = psi fragments.
// C layout then puts the 16 output-state amplitudes of one sample in the
// 8 accumulator rows x lane pair (L, L+16):
//   Z0 sign = n bit3 = (lane>=16)  -> uniform per lane,
//   Z2 sign = n bit1 = (row v & 2) -> compile-time per row,
// so the Z reduction is in-register adds + ONE shfl_xor(16) each.
// Lanes 0-15 store Z0, lanes 16-31 store Z2: zero divergence.
// 256 threads = 8 waves, one 16-sample tile per wave.
// ---------------------------------------------------------------------------
__global__ __launch_bounds__(256) void qnn_apply_kernel(
    const float* __restrict__ x,        // [B,4]
    const float* __restrict__ bfrag,    // 16*512 floats
    float* __restrict__ out,            // [B,32]
    int nTiles)
{
  const int lane = threadIdx.x & 31;
  const int wave = threadIdx.x >> 5;
  const int tile = blockIdx.x * 8 + wave;
  if (tile >= nTiles) return;

  // Pull the 32 KB fragment table toward the caches (global_prefetch_b8).
  {
    const char* base = (const char*)bfrag;
    #pragma unroll
    for (int i = 0; i < 8; ++i)
      __builtin_prefetch(base + (size_t)(i * 32 + lane) * 128, 0, 0);
  }

  const int s  = lane & 15;     // sample this lane feeds (B-matrix N / C N)
  const int hi = lane >> 4;     // K offset: lanes>=16 hold K = v+2
  const int b  = tile * 16 + s;

  // --- RX angle embedding of normalized x: psi[k] = (-i)^popc(k) * prod ---
  const v4f xs = ((const v4f*)x)[b];
  const float inv = rsqrtf(xs.x*xs.x + xs.y*xs.y + xs.z*xs.z + xs.w*xs.w) * 0.5f;
  const float hv[4] = { xs.x*inv, xs.y*inv, xs.z*inv, xs.w*inv };
  float cw[4], sw[4];
  #pragma unroll
  for (int w = 0; w < 4; ++w) __sincosf(hv[w], &sw[w], &cw[w]);

  float psr[8], psi[8];         // the 8 K-values this lane owns (j = kc*2 + v)
  #pragma unroll
  for (int j = 0; j < 8; ++j) {
    const int kc = j >> 1, v = j & 1;
    const int k  = 4*kc + v + 2*hi;
    const float prod = ((k & 8) ? sw[0] : cw[0]) * ((k & 4) ? sw[1] : cw[1])
                     * ((k & 2) ? sw[2] : cw[2]) * ((k & 1) ? sw[3] : cw[3]);
    const int pc = __popc(k);
    if (pc & 1) { psr[j] = 0.f; psi[j] = (pc & 2) ?  prod : -prod; }
    else        { psi[j] = 0.f; psr[j] = (pc & 2) ? -prod :  prod; }
  }

  // Hoisted B-fragments (constant across the p loop).
  v2f br[4], bi[4], bn[4];
  #pragma unroll
  for (int kc = 0; kc < 4; ++kc) {
    br[kc] = (v2f){  psr[2*kc],  psr[2*kc+1] };
    bi[kc] = (v2f){  psi[2*kc],  psi[2*kc+1] };
    bn[kc] = (v2f){ -psi[2*kc], -psi[2*kc+1] };
  }

  const float s0l = hi ? -1.f : 1.f;          // Z0 sign: n bit3 == (lane>=16)
  const int   obase = b * 32 + hi * 16;       // lanes<16 -> Z0 cols, >=16 -> Z2

  #pragma unroll
  for (int pg = 0; pg < 4; ++pg) {            // 4 groups of 4 p -> b128 stores
    float vals[4];
    #pragma unroll
    for (int pj = 0; pj < 4; ++pj) {
      const int p = pg * 4 + pj;
      const v4f* ap = (const v4f*)(bfrag + p*512 + lane*16);

      v8f Cr = {0.f,0.f,0.f,0.f,0.f,0.f,0.f,0.f};
      v8f Ci = {0.f,0.f,0.f,0.f,0.f,0.f,0.f,0.f};

      // Complex GEMM: Cr = Ur*Pr - Ui*Pi ; Ci = Ur*Pi + Ui*Pr
      #pragma unroll
      for (int kc = 0; kc < 4; ++kc) {
        const v4f f = ap[kc];
        const v2f aur = (v2f){ f.x, f.y };    // Ur(K=v0), Ur(K=v1)
        const v2f aui = (v2f){ f.z, f.w };    // Ui(K=v0), Ui(K=v1)
        Cr = __builtin_amdgcn_wmma_f32_16x16x4_f32(false, aur, false, br[kc], (short)0, Cr, false, false);
        Cr = __builtin_amdgcn_wmma_f32_16x16x4_f32(false, aui, false, bn[kc], (short)0, Cr, false, false);
        Ci = __builtin_amdgcn_wmma_f32_16x16x4_f32(false, aur, false, bi[kc], (short)0, Ci, false, false);
        Ci = __builtin_amdgcn_wmma_f32_16x16x4_f32(false, aui, false, br[kc], (short)0, Ci, false, false);
      }

      // probs per output-state row; n = v + 8*hi.
      float pr[8];
      #pragma unroll
      for (int v = 0; v < 8; ++v) pr[v] = Cr[v]*Cr[v] + Ci[v]*Ci[v];

      const float zp = ((pr[0]+pr[1]) + (pr[2]+pr[3]))
                     + ((pr[4]+pr[5]) + (pr[6]+pr[7]));
      float z2 = ((pr[0]+pr[1]) - (pr[2]+pr[3]))        // Z2 sign = (v&2)
               + ((pr[4]+pr[5]) - (pr[6]+pr[7]));
      float z0 = s0l * zp;
      z0 += __shfl_xor(z0, 16, 32);                     // combine lane halves
      z2 += __shfl_xor(z2, 16, 32);
      vals[pj] = hi ? z2 : z0;
    }
    *(v4f*)(out + obase + pg*4) = (v4f){ vals[0], vals[1], vals[2], vals[3] };
  }
}

// ---------------------------------------------------------------------------
extern "C" void kernel_launch(void* const* d_in, const int* in_sizes, int n_in,
                              void* d_out, int out_size, void* d_ws, size_t ws_size,
                              hipStream_t stream) {
  const float* x          = (const float*)d_in[0];   // [B,4]
  const float* p_rotation = (const float*)d_in[1];   // [16,12]
  const float* p_ising    = (const float*)d_in[2];   // [16,4]
  const float* p_entangle = (const float*)d_in[3];   // [16,12]
  float* out  = (float*)d_out;
  float* frag = (float*)d_ws;                        // 16*512 floats = 32 KB

  const int B = in_sizes[0] / 4;
  const int nTiles = B / 16;                         // 2048 for B=32768

  qnn_build_kernel<<<16, 32, 0, stream>>>(p_rotation, p_ising, p_entangle, frag);
  qnn_apply_kernel<<<(nTiles + 7) / 8, 256, 0, stream>>>(x, frag, out, nTiles);
}